// EGNN_4973572129358
// MI455X (gfx1250) — compile-verified
//
#include <hip/hip_runtime.h>
#include <hip/hip_bf16.h>

#define N_NODES 100000
#define N_EDGES 1600000
#define HCH     32
#define NGRAPH  512
#define F_INDIM 7

static __device__ __forceinline__ float rrelu_f(float v) {
    const float slope = (1.0f / 8.0f + 1.0f / 3.0f) * 0.5f;
    return v >= 0.0f ? v : slope * v;
}

// float atomic max that is correct for mixed signs:
// non-negative values -> signed-int max; negative values -> unsigned-int min.
static __device__ __forceinline__ void atomicMaxF(float* addr, float value) {
    if (value >= 0.0f) {
        atomicMax((int*)addr, __float_as_int(value));
    } else {
        atomicMin((unsigned int*)addr, __float_as_uint(value));
    }
}

// ---------------- init kernels ----------------

__global__ void k_zero(float* __restrict__ p, int n) {
    int i = blockIdx.x * blockDim.x + threadIdx.x;
    if (i < n) p[i] = 0.0f;
}

// out[n*32+k] = b[k]  (scatter-add then lands on top of the bias)
__global__ void k_bias_init(float* __restrict__ out, const float* __restrict__ b, int total) {
    int i = blockIdx.x * blockDim.x + threadIdx.x;
    if (i < total) out[i] = b[i & (HCH - 1)];
}

// ---------------- degree / norm ----------------

__global__ void k_degrees(const int* __restrict__ ei, const float* __restrict__ ew,
                          float* __restrict__ deg1, float* __restrict__ deg2) {
    int e = blockIdx.x * blockDim.x + threadIdx.x;
    if (e < N_EDGES) {
        int col = ei[N_EDGES + e];
        atomicAdd(&deg1[col], ew[e]);
        atomicAdd(&deg2[col], 1.0f);
    }
}

__global__ void k_rsqrt_inplace(float* __restrict__ p, int n) {
    int i = blockIdx.x * blockDim.x + threadIdx.x;
    if (i < n) {
        float v = p[i];
        p[i] = (v > 0.0f) ? rsqrtf(v) : 0.0f;
    }
}

// ---------------- conv1: h1 = x @ W1 (K = 7) ----------------

__global__ void k_linear1(const float* __restrict__ x, const float* __restrict__ W1,
                          float* __restrict__ h1) {
    int i = blockIdx.x * blockDim.x + threadIdx.x;
    if (i >= N_NODES * HCH) return;
    int n = i >> 5, k = i & 31;
    float s = 0.0f;
#pragma unroll
    for (int j = 0; j < F_INDIM; ++j)
        s += x[n * F_INDIM + j] * W1[j * HCH + k];
    h1[i] = s;
}

// ---------------- edge scatter: out[col] += dis[row]*ew*dis[col] * h[row] ----------------
// one lane per (edge, channel): channel dim is contiguous & coalesced,
// per-edge scalars are same-address wave broadcasts.

__global__ void k_scatter_w(const int* __restrict__ ei, const float* __restrict__ ew,
                            const float* __restrict__ dis, const float* __restrict__ h,
                            float* __restrict__ out) {
    long long tid = (long long)blockIdx.x * blockDim.x + threadIdx.x;
    int e = (int)(tid >> 5);
    int k = (int)(tid & 31);
    if (e >= N_EDGES) return;
    int row = ei[e];
    int col = ei[N_EDGES + e];
    float norm = dis[row] * ew[e] * dis[col];
    atomicAdd(&out[col * HCH + k], norm * h[row * HCH + k]);
}

__global__ void k_scatter_u(const int* __restrict__ ei, const float* __restrict__ dis,
                            const float* __restrict__ h, float* __restrict__ out) {
    long long tid = (long long)blockIdx.x * blockDim.x + threadIdx.x;
    int e = (int)(tid >> 5);
    int k = (int)(tid & 31);
    if (e >= N_EDGES) return;
    int row = ei[e];
    int col = ei[N_EDGES + e];
    float norm = dis[row] * dis[col];
    atomicAdd(&out[col * HCH + k], norm * h[row * HCH + k]);
}

// ---------------- neighbor max pool: xp[col] = max(xp[col], src[row]) ----------------

__global__ void k_maxpool_edges(const int* __restrict__ ei, const float* __restrict__ src,
                                float* __restrict__ xp) {
    long long tid = (long long)blockIdx.x * blockDim.x + threadIdx.x;
    int e = (int)(tid >> 5);
    int k = (int)(tid & 31);
    if (e >= N_EDGES) return;
    int row = ei[e];
    int col = ei[N_EDGES + e];
    atomicMaxF(&xp[col * HCH + k], src[row * HCH + k]);
}

// ---------------- WMMA GEMM: C[N,32] = A[N,32] @ B[32,32], f32 exact ----------------
// V_WMMA_F32_16X16X4_F32, K stepped by 4. One 16x16 tile per wave.
// Layouts (ISA 7.12.2): lane L -> m=L%16, h=L/16.
//   A comp v : A[m][k0 + v + 2h]      (16x4, 2 VGPRs)
//   B comp v : B[k0 + v + 2h][n]      (4x16, 2 VGPRs)
//   D VGPR r : C[r + 8h][m]           (16x16, 8 VGPRs)

typedef __attribute__((ext_vector_type(2))) float v2f;
typedef __attribute__((ext_vector_type(8))) float v8f;

__global__ __launch_bounds__(256) void k_gemm32_wmma(const float* __restrict__ A,
                                                     const float* __restrict__ B,
                                                     float* __restrict__ C, int nrows) {
    int wave = threadIdx.x >> 5;                 // 0..7
    int lane = threadIdx.x & 31;
    int rtile = blockIdx.x * 4 + (wave >> 1);    // 16-row tile
    int ctile = wave & 1;                        // 16-col tile (HCH=32 -> 2 tiles)
    int row0 = rtile * 16;
    if (row0 >= nrows) return;                   // wave-uniform guard: EXEC stays all-1s
    int m = lane & 15;
    int h = lane >> 4;

    v8f acc = {};
#pragma unroll
    for (int k0 = 0; k0 < HCH; k0 += 4) {
        v2f a, b;
        a.x = A[(row0 + m) * HCH + k0 + 2 * h + 0];
        a.y = A[(row0 + m) * HCH + k0 + 2 * h + 1];
        b.x = B[(k0 + 2 * h + 0) * HCH + ctile * 16 + m];
        b.y = B[(k0 + 2 * h + 1) * HCH + ctile * 16 + m];
        acc = __builtin_amdgcn_wmma_f32_16x16x4_f32(
            /*neg_a=*/false, a, /*neg_b=*/false, b,
            /*c_mod=*/(short)0, acc, /*reuse_a=*/false, /*reuse_b=*/false);
    }
#pragma unroll
    for (int r = 0; r < 8; ++r)
        C[(row0 + r + 8 * h) * HCH + ctile * 16 + m] = acc[r];
}

// ---------------- relu(res + conv2) fused with global max pool ----------------
// relu output >= 0 and gm is zero-initialized, so signed-int atomicMax is exact.

__global__ void k_relu_gmax(const float* __restrict__ xp, const float* __restrict__ out2,
                            const int* __restrict__ batch, float* __restrict__ gm) {
    int i = blockIdx.x * blockDim.x + threadIdx.x;
    if (i >= N_NODES * HCH) return;
    float v = xp[i] + out2[i];
    v = v > 0.0f ? v : 0.0f;
    int g = batch[i >> 5];
    atomicMax((int*)&gm[g * HCH + (i & 31)], __float_as_int(v));
}

// ---------------- MLP head ----------------

__global__ void k_mlp_res(const float* __restrict__ in, const float* __restrict__ W,
                          const float* __restrict__ b, float* __restrict__ out) {
    int i = blockIdx.x * blockDim.x + threadIdx.x;
    if (i >= NGRAPH * HCH) return;
    int g = i >> 5, k = i & 31;
    float t = b[k];
#pragma unroll
    for (int j = 0; j < HCH; ++j)
        t += in[g * HCH + j] * W[j * HCH + k];
    out[i] = rrelu_f(in[i] + t);
}

__global__ void k_final(const float* __restrict__ in, const float* __restrict__ Wl3,
                        const float* __restrict__ bl3, float* __restrict__ out) {
    int g = blockIdx.x * blockDim.x + threadIdx.x;
    if (g >= NGRAPH) return;
    float t = bl3[0];
#pragma unroll
    for (int j = 0; j < HCH; ++j)
        t += in[g * HCH + j] * Wl3[j];
    out[g] = rrelu_f(t);
}

// ---------------- launcher ----------------

extern "C" void kernel_launch(void* const* d_in, const int* in_sizes, int n_in,
                              void* d_out, int out_size, void* d_ws, size_t ws_size,
                              hipStream_t stream) {
    const float* x     = (const float*)d_in[0];
    const int*   ei    = (const int*)d_in[1];     // [2,E] flat
    const int*   batch = (const int*)d_in[2];
    const float* ew    = (const float*)d_in[3];
    const float* W1    = (const float*)d_in[4];
    const float* b1    = (const float*)d_in[5];
    const float* W2    = (const float*)d_in[6];
    const float* b2    = (const float*)d_in[7];
    const float* Wl1   = (const float*)d_in[8];
    const float* bl1   = (const float*)d_in[9];
    const float* Wl2   = (const float*)d_in[10];
    const float* bl2   = (const float*)d_in[11];
    const float* Wl3   = (const float*)d_in[12];
    const float* bl3   = (const float*)d_in[13];
    float* outp        = (float*)d_out;

    float* w = (float*)d_ws;
    float* deg1 = w; w += N_NODES;               // -> dis1 in place
    float* deg2 = w; w += N_NODES;               // -> dis2 in place
    float* h1   = w; w += (size_t)N_NODES * HCH;
    float* out1 = w; w += (size_t)N_NODES * HCH;
    float* xp   = w; w += (size_t)N_NODES * HCH;
    float* h2   = w; w += (size_t)N_NODES * HCH;
    float* out2 = w; w += (size_t)N_NODES * HCH;
    float* gm   = w; w += (size_t)NGRAPH * HCH;
    float* m1   = w; w += (size_t)NGRAPH * HCH;
    float* m2   = w; w += (size_t)NGRAPH * HCH;

    const int B = 256;
    const long long elN  = (long long)N_NODES * HCH;
    const long long elEK = (long long)N_EDGES * HCH;
    dim3 gN((unsigned)((elN + B - 1) / B));
    dim3 gEK((unsigned)((elEK + B - 1) / B));
    dim3 gE((N_EDGES + B - 1) / B);
    dim3 gNode((N_NODES + B - 1) / B);

    // init
    k_zero<<<gNode, B, 0, stream>>>(deg1, N_NODES);
    k_zero<<<gNode, B, 0, stream>>>(deg2, N_NODES);
    k_zero<<<(NGRAPH * HCH + B - 1) / B, B, 0, stream>>>(gm, NGRAPH * HCH);
    k_bias_init<<<gN, B, 0, stream>>>(out1, b1, N_NODES * HCH);
    k_bias_init<<<gN, B, 0, stream>>>(out2, b2, N_NODES * HCH);

    // degrees -> rsqrt norms (in place)
    k_degrees<<<gE, B, 0, stream>>>(ei, ew, deg1, deg2);
    k_rsqrt_inplace<<<gNode, B, 0, stream>>>(deg1, N_NODES);
    k_rsqrt_inplace<<<gNode, B, 0, stream>>>(deg2, N_NODES);

    // conv1
    k_linear1<<<gN, B, 0, stream>>>(x, W1, h1);
    k_scatter_w<<<gEK, B, 0, stream>>>(ei, ew, deg1, h1, out1);

    // neighbor max pool (self init via d2d copy, then edge atomic-max)
    hipMemcpyAsync(xp, out1, (size_t)N_NODES * HCH * sizeof(float),
                   hipMemcpyDeviceToDevice, stream);
    k_maxpool_edges<<<gEK, B, 0, stream>>>(ei, out1, xp);

    // conv2: WMMA GEMM h2 = xp @ W2, then scatter with unit edge weights
    {
        int rtiles = N_NODES / 16;               // 6250, exact
        int blocks = (rtiles + 3) / 4;
        k_gemm32_wmma<<<blocks, 256, 0, stream>>>(xp, W2, h2, N_NODES);
    }
    k_scatter_u<<<gEK, B, 0, stream>>>(ei, deg2, h2, out2);

    // relu(res + conv2) + global max pool
    k_relu_gmax<<<gN, B, 0, stream>>>(xp, out2, batch, gm);

    // MLP head
    k_mlp_res<<<(NGRAPH * HCH + B - 1) / B, B, 0, stream>>>(gm, Wl1, bl1, m1);
    k_mlp_res<<<(NGRAPH * HCH + B - 1) / B, B, 0, stream>>>(m1, Wl2, bl2, m2);
    k_final<<<(NGRAPH + B - 1) / B, B, 0, stream>>>(m2, Wl3, bl3, outp);
}